// CustomConv_73289321939211
// MI455X (gfx1250) — compile-verified
//
#include <hip/hip_runtime.h>

// CustomConv on gfx1250 (MI455X):
//   Pass 1: per-region 15x15 cross-correlation as an implicit Toeplitz GEMM on
//           V_WMMA_F32_16X16X4_F32.  K is enumerated k = v*32 + yy so that both
//           WMMA operands are *linear* LDS loads:
//              A[m][k] = KtZ[16 + k - m]   (zero-guarded banded kernel matrix)
//              B[k][n] = PtT[k + 32*n]     (transposed zero-padded image tile)
//           -> inner loop is 4x ds_load_b32 + 1x v_wmma, no VALU address math.
//   Pass 2: 16-tap row box-sum  d_out -> d_ws.
//   Pass 3: 16-tap col box-sum  d_ws  -> d_out   (final result).
// Scratch requirement: 12*256*256*4 = 3,145,728 bytes of d_ws.

typedef float v2f __attribute__((ext_vector_type(2)));
typedef float v8f __attribute__((ext_vector_type(8)));

#define KS   15            // kernel size
#define RG   16            // region size
#define HW   256           // image H=W
#define HO   241           // sliding positions per axis (256-16+1)
#define PADI 7             // (KS-1)/2
#define TILE 30            // 16 + 15 - 1 padded-tile extent
#define YS   32            // per-v row-stride / padded yy extent
#define GK   480           // GEMM K extent = 15 * 32
#define AGUARD 16          // zero guard in front of KtZ (covers k - m < 0)

__global__ __launch_bounds__(32)
void CustomConv_corr_wmma(const float* __restrict__ image,
                          const float* __restrict__ kernels,
                          float* __restrict__ corr)
{
    __shared__ float PtT[TILE * YS];        // transposed tile: PtT[x*32 + y]
    __shared__ float KtZ[AGUARD + KS * YS]; // guarded kernel:  KtZ[16 + v*32 + u]

    const int r  = blockIdx.x;              // region id = (i/16)*16 + (j/16)
    const int bc = blockIdx.y;              // b*3 + c
    const int i0 = (r >> 4) * RG;
    const int j0 = (r & 15) * RG;
    const int l  = threadIdx.x;             // 0..31 (wave32)

    // ---- zero-fill LDS (guard zones, band padding, rows 30/31) ----
    for (int e = l; e < TILE * YS; e += 32)          PtT[e] = 0.0f;
    for (int e = l; e < AGUARD + KS * YS; e += 32)   KtZ[e] = 0.0f;

    // ---- stage kernel, column-major with stride 32 ----
    for (int e = l; e < KS * KS; e += 32) {
        int v = e / KS, u = e - v * KS;              // setup-only division
        KtZ[AGUARD + v * YS + u] = kernels[r * (KS * KS) + u * KS + v];
    }

    // ---- stage zero-padded 30x30 image tile, transposed ----
    const float* img = image + bc * (HW * HW);
    for (int e = l; e < TILE * TILE; e += 32) {
        int yy = e / TILE, xx = e - yy * TILE;       // consecutive lanes -> consecutive gx
        int gy = i0 + yy - PADI, gx = j0 + xx - PADI;
        float pv = 0.0f;
        if (gy >= 0 && gy < HW && gx >= 0 && gx < HW)
            pv = img[gy * HW + gx];
        PtT[xx * YS + yy] = pv;
    }
    __syncthreads();

    const int m  = l & 15;                  // A row (= i offset) / B,D column (= j offset)
    const int lh = l >> 4;                  // lane-half: k sub-pair selector (ISA layout)

    // Per-lane float-index bases; per step the address is base + k0 + p.
    const float* aBase = KtZ + (AGUARD + 2 * lh - m);   // >= KtZ + 1, always in-bounds
    const float* bBase = PtT + (YS * m + 2 * lh);       // max idx 32*15+2+479 = 959

    // corr[i0+m, j0+n] = sum_{u,v} ker[u,v] * P[i0+m+u, j0+n+v]
    //                  = sum_k KtZ[16+k-m] * PtT[k+32n],  k = v*32 + yy
    v8f acc = {};
    #pragma unroll 8
    for (int k0 = 0; k0 < GK; k0 += 4) {    // 120 WMMA steps
        v2f a, b;
        a[0] = aBase[k0];     a[1] = aBase[k0 + 1];
        b[0] = bBase[k0];     b[1] = bBase[k0 + 1];
        acc = __builtin_amdgcn_wmma_f32_16x16x4_f32(
                  false, a, false, b, (short)0, acc, false, false);
    }

    // ---- store masked corr tile (zeros beyond 241x241 valid grid) ----
    float* dst = corr + bc * (HW * HW);
    #pragma unroll
    for (int g = 0; g < 8; ++g) {
        int mi = lh * 8 + g;                // ISA 16x16 f32 C/D layout: M = 8*half + g
        int i = i0 + mi, j = j0 + m;
        float ov = (i < HO && j < HO) ? acc[g] : 0.0f;
        dst[i * HW + j] = ov;
    }
}

__global__ __launch_bounds__(256)
void CustomConv_rowbox(const float* __restrict__ src, float* __restrict__ dst)
{
    int idx = blockIdx.x * 256 + threadIdx.x;        // bc*65536 + y*256 + x
    int x = idx & 255;
    float s = 0.0f;
    #pragma unroll
    for (int d = 0; d < RG; ++d) {
        float v = (x - d >= 0) ? src[idx - d] : 0.0f;
        s += v;
    }
    dst[idx] = s;
}

__global__ __launch_bounds__(256)
void CustomConv_colbox(const float* __restrict__ src, float* __restrict__ dst)
{
    int idx = blockIdx.x * 256 + threadIdx.x;
    int y = (idx >> 8) & 255;
    float s = 0.0f;
    #pragma unroll
    for (int d = 0; d < RG; ++d) {
        float v = (y - d >= 0) ? src[idx - d * HW] : 0.0f;
        s += v;
    }
    dst[idx] = s;
}

extern "C" void kernel_launch(void* const* d_in, const int* in_sizes, int n_in,
                              void* d_out, int out_size, void* d_ws, size_t ws_size,
                              hipStream_t stream)
{
    const float* image   = (const float*)d_in[0];   // [4,3,256,256] f32
    const float* kernels = (const float*)d_in[1];   // [256,1,15,15] f32
    float* out = (float*)d_out;                     // [4,3,256,256] f32
    float* ws  = (float*)d_ws;                      // >= 3.15 MB scratch

    dim3 g1(256, 12);                               // (region, b*3+c)
    CustomConv_corr_wmma<<<g1, 32, 0, stream>>>(image, kernels, out);

    const int total = 12 * HW * HW;                 // 786432
    CustomConv_rowbox<<<total / 256, 256, 0, stream>>>(out, ws);
    CustomConv_colbox<<<total / 256, 256, 0, stream>>>(ws, out);
}